// TreeLayer_34626026340906
// MI455X (gfx1250) — compile-verified
//
#include <hip/hip_runtime.h>
#include <cstdint>

// ---------------- problem constants (match reference) ----------------
#define Q_DIMc   8
#define N_HEADc  4
#define DEPTHc   8
#define V_DIMc   8
#define NVc      256
#define NKc      255
#define KEY_FLOATS (N_HEADc * NKc * 2 * Q_DIMc)   // 16320 floats = 65280 B (fits 64KB LDS)

struct TfKeys {
  uint32_t kb0[DEPTHc], kb1[DEPTHc];   // per-depth kbits keys
  uint32_t kl0[DEPTHc], kl1[DEPTHc];   // per-depth lor keys
  uint32_t kv0, kv1;                   // value key
};

// ---------------- threefry2x32 (exact JAX algorithm) ----------------
__host__ __device__ __forceinline__ uint32_t tf_rotl(uint32_t x, int r) {
  return (x << r) | (x >> (32 - r));
}
__host__ __device__ __forceinline__ void tf2x32(uint32_t k0, uint32_t k1,
                                                uint32_t x0, uint32_t x1,
                                                uint32_t& o0, uint32_t& o1) {
  const uint32_t ks2 = k0 ^ k1 ^ 0x1BD11BDAu;
  x0 += k0; x1 += k1;
#define TF_R(r) { x0 += x1; x1 = tf_rotl(x1, r); x1 ^= x0; }
  TF_R(13) TF_R(15) TF_R(26) TF_R(6)   x0 += k1;  x1 += ks2 + 1u;
  TF_R(17) TF_R(29) TF_R(16) TF_R(24)  x0 += ks2; x1 += k0  + 2u;
  TF_R(13) TF_R(15) TF_R(26) TF_R(6)   x0 += k0;  x1 += k1  + 3u;
  TF_R(17) TF_R(29) TF_R(16) TF_R(24)  x0 += k1;  x1 += ks2 + 4u;
  TF_R(13) TF_R(15) TF_R(26) TF_R(6)   x0 += ks2; x1 += k0  + 5u;
#undef TF_R
  o0 = x0; o1 = x1;
}

// ---------------- device math helpers ----------------
__device__ __forceinline__ float tf_u01(uint32_t b) {
  // JAX f32 uniform: ((bits>>9)|0x3f800000) bitcast - 1.0
  return __uint_as_float((b >> 9) | 0x3f800000u) - 1.0f;
}
__device__ __forceinline__ float sigm(float x) { return 1.0f / (1.0f + __expf(-x)); }
__device__ __forceinline__ float laddexp(float a, float b) {  // jnp.logaddexp
  const float m = fmaxf(a, b);
  return m + log1pf(__expf(-fabsf(a - b)));
}

// ---------------- kernel ----------------
// One thread handles sibling rows (t, t+BH/2): JAX threefry pairs counters
// (i, i+n/2), so both 32-bit hash outputs are consumed -> 2x less hashing.
__global__ __launch_bounds__(256) void treelayer_kernel(
    const uint8_t* __restrict__ qb, const float* __restrict__ qs,
    const float* __restrict__ key_p, const float* __restrict__ value_p,
    float* __restrict__ out, int BH, TfKeys K)
{
  extern __shared__ float skey[];   // 16320 floats: key_p staged per workgroup

  const int t = (int)(blockIdx.x * blockDim.x + threadIdx.x);
  const int halfBH = BH >> 1;

  // Warm L2/WGP$ for this thread's qs rows while the TDM runs (global_prefetch_b8).
  if (t < halfBH) {
    __builtin_prefetch(qs + (size_t)t * 8, 0, 3);
    __builtin_prefetch(qs + (size_t)(t + halfBH) * 8, 0, 3);
  }

  // ---- Stage key table into LDS via CDNA5 Tensor Data Mover (async DMA) ----
#if defined(__HIP_DEVICE_COMPILE__) && defined(__gfx1250__) && __has_builtin(__builtin_amdgcn_tensor_load_to_lds)
  if (threadIdx.x < 32u) {  // wave 0 issues one TDM descriptor; EXEC-independent DMA
    typedef unsigned int v4u __attribute__((ext_vector_type(4)));
    typedef int          v8i __attribute__((ext_vector_type(8)));
    typedef int          v4i __attribute__((ext_vector_type(4)));
    const uint64_t ga  = (uint64_t)(uintptr_t)key_p;
    const uint32_t lds = (uint32_t)(uintptr_t)(&skey[0]);   // flat low32 == LDS offset
    const uint32_t n   = (uint32_t)KEY_FLOATS;              // 4-byte elements
    v4u g0;
    g0[0] = 1u;                                             // count=1, user descriptor
    g0[1] = lds;                                            // lds_addr (bytes)
    g0[2] = (uint32_t)(ga & 0xFFFFFFFFu);                   // global_addr[31:0]
    g0[3] = (uint32_t)((ga >> 32) & 0x01FFFFFFu) | (2u << 30); // addr[56:32] | type=2
    v8i g1;
    g1[0] = (int)(2u << 16);                                // wg_mask=0, data_size=4B
    g1[1] = (int)((n & 0xFFFFu) << 16);                     // tensor_dim0[15:0]
    g1[2] = (int)(((n >> 16) & 0xFFFFu) | (1u << 16));      // dim0[31:16] | tensor_dim1=1
    g1[3] = (int)((n & 0xFFFFu) << 16);                     // tile_dim0 = n (1-D tile)
    g1[4] = (int)(1u | (1u << 16));                         // tile_dim1=1, tile_dim2=1
    g1[5] = (int)n;                                         // tensor_dim0_stride
    g1[6] = (int)((n & 0xFFFFu) << 16);                     // dim1_stride lo16
    g1[7] = 0;
    v4i g2; g2[0] = 1; g2[1] = 1; g2[2] = 0; g2[3] = (int)(1u << 16); // dim2=1,dim3=1,tile3=1
    v4i g3; g3[0] = 0; g3[1] = (int)(1u << 16); g3[2] = (int)(1u << 16); g3[3] = 0;
#if __clang_major__ >= 23
    v8i gz = {0, 0, 0, 0, 0, 0, 0, 0};
    __builtin_amdgcn_tensor_load_to_lds(g0, g1, g2, g3, gz, 0);
#else
    __builtin_amdgcn_tensor_load_to_lds(g0, g1, g2, g3, 0);
#endif
    __builtin_amdgcn_s_wait_tensorcnt(0);                   // s_wait_tensorcnt 0
  }
#else
  for (int i = (int)threadIdx.x; i < KEY_FLOATS / 4; i += (int)blockDim.x)
    reinterpret_cast<float4*>(skey)[i] = reinterpret_cast<const float4*>(key_p)[i];
#endif
  __syncthreads();

  if (t >= halfBH) return;

  const int rA = t, rB = t + halfBH;
  const int h  = rA & (N_HEADc - 1);          // rB has same head (halfBH % 4 == 0)
  const int bA = rA >> 2, bB = rB >> 2;

  // qs_eff = (2*q - 1) * qs
  float qeA[8], qeB[8];
  {
    const float4* qs4 = reinterpret_cast<const float4*>(qs);
    float4 lo = qs4[(size_t)rA * 2], hi = qs4[(size_t)rA * 2 + 1];
    qeA[0]=lo.x; qeA[1]=lo.y; qeA[2]=lo.z; qeA[3]=lo.w;
    qeA[4]=hi.x; qeA[5]=hi.y; qeA[6]=hi.z; qeA[7]=hi.w;
    lo = qs4[(size_t)rB * 2]; hi = qs4[(size_t)rB * 2 + 1];
    qeB[0]=lo.x; qeB[1]=lo.y; qeB[2]=lo.z; qeB[3]=lo.w;
    qeB[4]=hi.x; qeB[5]=hi.y; qeB[6]=hi.z; qeB[7]=hi.w;
#pragma unroll
    for (int j = 0; j < 8; ++j) {
      qeA[j] = qb[(size_t)bA * 8 + j] ? qeA[j] : -qeA[j];
      qeB[j] = qb[(size_t)bB * 8 + j] ? qeB[j] : -qeB[j];
    }
  }

  int   ixA = 0,  ixB = 0;
  float supA = 0.f, supB = 0.f;

#pragma unroll 1                       // keep ~10KB loop body; inner loops unrolled
  for (int d = 0; d < DEPTHc; ++d) {
    const uint32_t kb0 = K.kb0[d], kb1 = K.kb1[d];
    const int lvl = (1 << d) - 1 + h * NKc;
    const float* rowA = skey + (size_t)(lvl + ixA) * 16;
    const float* rowB = skey + (size_t)(lvl + ixB) * 16;
    const uint32_t fA    = (uint32_t)t * 16u;        // flat kbits index of row A
    const uint32_t dHalf = (uint32_t)halfBH * 16u;   // n/2 for shape (BH,16)

    float tA[8], tB[8];
#pragma unroll
    for (int j = 0; j < 8; ++j) {
      const float2 wA = *reinterpret_cast<const float2*>(rowA + 2 * j); // ds_load_b64
      const float2 wB = *reinterpret_cast<const float2*>(rowB + 2 * j);
      uint32_t u0, u1;
      tf2x32(kb0, kb1, fA + 2u * j, fA + 2u * j + dHalf, u0, u1);
      const float bA0 = (tf_u01(u0) < sigm(wA.x)) ? 1.f : 0.f;
      const float bB0 = (tf_u01(u1) < sigm(wB.x)) ? 1.f : 0.f;
      tf2x32(kb0, kb1, fA + 2u * j + 1u, fA + 2u * j + 1u + dHalf, u0, u1);
      const float bA1 = (tf_u01(u0) < sigm(wA.y)) ? 1.f : 0.f;
      const float bB1 = (tf_u01(u1) < sigm(wB.y)) ? 1.f : 0.f;
      const float k1A = -laddexp(-bA0,  qeA[j]);
      const float k2A = -laddexp(-bA1, -qeA[j]);
      tA[j] = laddexp(k1A, k2A);
      const float k1B = -laddexp(-bB0,  qeB[j]);
      const float k2B = -laddexp(-bB1, -qeB[j]);
      tB[j] = laddexp(k1B, k2B);
    }

    // lor_s = -logsumexp(-t)  (per-thread register reduction; no cross-lane needed)
    float mnA = tA[0], mnB = tB[0];
#pragma unroll
    for (int j = 1; j < 8; ++j) { mnA = fminf(mnA, tA[j]); mnB = fminf(mnB, tB[j]); }
    float sA = 0.f, sB = 0.f;
#pragma unroll
    for (int j = 0; j < 8; ++j) { sA += __expf(mnA - tA[j]); sB += __expf(mnB - tB[j]); }
    float lsA = mnA - __logf(sA);
    float lsB = mnB - __logf(sB);

    uint32_t u0, u1;
    tf2x32(K.kl0[d], K.kl1[d], (uint32_t)t, (uint32_t)(t + halfBH), u0, u1);
    const int lA = tf_u01(u0) < sigm(lsA);
    const int lB = tf_u01(u1) < sigm(lsB);
    lsA = lA ? lsA : -lsA;
    lsB = lB ? lsB : -lsB;
    ixA = 2 * ixA + lA;
    ixB = 2 * ixB + lB;
    supA = (d == 0) ? lsA : -laddexp(-supA, -lsA);
    supB = (d == 0) ? lsB : -laddexp(-supB, -lsB);
  }

  // ---- value phase: 32B gather from L2-resident table, sample, combine ----
  const float4* vp4 = reinterpret_cast<const float4*>(value_p);
  float vwA[8], vwB[8];
  {
    float4 lo = vp4[(size_t)(h * NVc + ixA) * 2], hi = vp4[(size_t)(h * NVc + ixA) * 2 + 1];
    vwA[0]=lo.x; vwA[1]=lo.y; vwA[2]=lo.z; vwA[3]=lo.w;
    vwA[4]=hi.x; vwA[5]=hi.y; vwA[6]=hi.z; vwA[7]=hi.w;
    lo = vp4[(size_t)(h * NVc + ixB) * 2]; hi = vp4[(size_t)(h * NVc + ixB) * 2 + 1];
    vwB[0]=lo.x; vwB[1]=lo.y; vwB[2]=lo.z; vwB[3]=lo.w;
    vwB[4]=hi.x; vwB[5]=hi.y; vwB[6]=hi.z; vwB[7]=hi.w;
  }
  float obA[8], osA[8], obB[8], osB[8];
  const uint32_t fv = (uint32_t)t * 8u;
  const uint32_t dv = (uint32_t)halfBH * 8u;   // n/2 for shape (BH,8)
#pragma unroll
  for (int v = 0; v < 8; ++v) {
    uint32_t u0, u1;
    tf2x32(K.kv0, K.kv1, fv + (uint32_t)v, fv + (uint32_t)v + dv, u0, u1);
    const int bv0 = tf_u01(u0) < sigm(vwA[v]);
    const int bv1 = tf_u01(u1) < sigm(vwB[v]);
    const float svA = bv0 ? vwA[v] : -vwA[v];
    const float svB = bv1 ? vwB[v] : -vwB[v];
    obA[v] = (float)bv0;  osA[v] = -laddexp(-svA, -supA);
    obB[v] = (float)bv1;  osB[v] = -laddexp(-svB, -supB);
  }

  float4* out4 = reinterpret_cast<float4*>(out);
  const size_t oA = (size_t)rA * 2, oB = (size_t)rB * 2;
  const size_t sOff = (size_t)BH * 2;   // value_s region, float4 units
  out4[oA]          = make_float4(obA[0], obA[1], obA[2], obA[3]);
  out4[oA + 1]      = make_float4(obA[4], obA[5], obA[6], obA[7]);
  out4[oB]          = make_float4(obB[0], obB[1], obB[2], obB[3]);
  out4[oB + 1]      = make_float4(obB[4], obB[5], obB[6], obB[7]);
  out4[sOff + oA]     = make_float4(osA[0], osA[1], osA[2], osA[3]);
  out4[sOff + oA + 1] = make_float4(osA[4], osA[5], osA[6], osA[7]);
  out4[sOff + oB]     = make_float4(osB[0], osB[1], osB[2], osB[3]);
  out4[sOff + oB + 1] = make_float4(osB[4], osB[5], osB[6], osB[7]);
}

// ---------------- host launcher ----------------
extern "C" void kernel_launch(void* const* d_in, const int* in_sizes, int n_in,
                              void* d_out, int out_size, void* d_ws, size_t ws_size,
                              hipStream_t stream) {
  const uint8_t* qb      = (const uint8_t*)d_in[0];
  const float*   qs      = (const float*)d_in[1];
  const float*   key_p   = (const float*)d_in[2];
  const float*   value_p = (const float*)d_in[3];
  float*         out     = (float*)d_out;
  const int BH = in_sizes[1] / Q_DIMc;   // qs is (B*H, Q)

  // Reproduce JAX key-split chain on host: rng=key(42)=(0,42);
  // per depth: rng,kb,kl = split(rng,3); finally rng,kv = split(rng,2).
  TfKeys K;
  uint32_t r0 = 0u, r1 = 42u;
  for (int d = 0; d < DEPTHc; ++d) {
    uint32_t a0, a1, b0, b1, c0, c1;
    tf2x32(r0, r1, 0u, 3u, a0, a1);   // split(3): hash i uses counters (i, i+3)
    tf2x32(r0, r1, 1u, 4u, b0, b1);   // bits = [a0,b0,c0,a1,b1,c1]
    tf2x32(r0, r1, 2u, 5u, c0, c1);
    r0 = a0; r1 = b0;                 // rng = (bits[0], bits[1])
    K.kb0[d] = c0; K.kb1[d] = a1;     // kb  = (bits[2], bits[3])
    K.kl0[d] = b1; K.kl1[d] = c1;     // kl  = (bits[4], bits[5])
  }
  {
    uint32_t a0, a1, b0, b1;
    tf2x32(r0, r1, 0u, 2u, a0, a1);   // split(2): bits = [a0,b0,a1,b1]
    tf2x32(r0, r1, 1u, 3u, b0, b1);
    K.kv0 = a1; K.kv1 = b1;           // kv = (bits[2], bits[3])
  }

  const int halfBH  = BH >> 1;
  const int threads = 256;                                  // 8 wave32 waves
  const int blocks  = (halfBH + threads - 1) / threads;     // 2048 for B=262144
  const size_t shmem = (size_t)KEY_FLOATS * sizeof(float);  // 65280 B < 64KB
  treelayer_kernel<<<blocks, threads, shmem, stream>>>(qb, qs, key_p, value_p, out, BH, K);
  (void)d_ws; (void)ws_size; (void)n_in; (void)out_size;
}